// GraphConv_16862041604212
// MI455X (gfx1250) — compile-verified
//
#include <hip/hip_runtime.h>

#define NN 100000
#define NE 1600000
#define D  64

typedef __attribute__((ext_vector_type(2))) float v2f;
typedef __attribute__((ext_vector_type(4))) float v4f;
typedef __attribute__((ext_vector_type(8))) float v8f;

// ---------------- kernel 1: zero workspace (h and in_deg) ----------------
__global__ void zero_ws_kernel(float* __restrict__ ws, int n) {
    int i = blockIdx.x * blockDim.x + threadIdx.x;
    if (i < n) ws[i] = 0.0f;
}

// ---------------- kernel 2: edge scatter: h[dst] += x[src] + edge_fea ----
// 16 threads per edge, 4 floats per thread (b128 loads, 4x f32 atomics).
__global__ void scatter_kernel(const float* __restrict__ x,
                               const float* __restrict__ edge_fea,
                               const int* __restrict__ src,
                               const int* __restrict__ dst,
                               float* __restrict__ h,
                               float* __restrict__ in_deg) {
    long tid = (long)blockIdx.x * blockDim.x + threadIdx.x;
    long e   = tid >> 4;
    if (e >= NE) return;
    int  q = (int)(tid & 15) * 4;
    int  s = src[e];
    int  t = dst[e];
    v4f xv = *(const v4f*)(x + (long)s * D + q);                       // L2-resident gather
    v4f ev = __builtin_nontemporal_load((const v4f*)(edge_fea + e * (long)D + q)); // streaming
    float* hp = h + (long)t * D + q;
    atomicAdd(hp + 0, xv.x + ev.x);
    atomicAdd(hp + 1, xv.y + ev.y);
    atomicAdd(hp + 2, xv.z + ev.z);
    atomicAdd(hp + 3, xv.w + ev.w);
    if ((tid & 15) == 0) atomicAdd(in_deg + t, 1.0f);
}

// ---------------- kernel 3: rst = (h @ W^T + b) * deg^-1/2 + x, via WMMA --
// One wave computes one 16x16 tile of rst. K=64 done as 16 steps of
// V_WMMA_F32_16X16X4_F32 (full fp32 accumulate, matches reference numerics).
// A (16x4 f32) ISA layout: lane l -> row M = l&15, VGPR0/1 hold K = khalf, khalf+1
// where khalf = (l>>4)*2.  B (4x16) mirrors the K split; B[k][j] = W[j][k] so
// each lane reads a contiguous float2 from row (j0+col) of W.
__global__ void gemm_norm_kernel(const float* __restrict__ h,
                                 const float* __restrict__ W,
                                 const float* __restrict__ bias,
                                 const float* __restrict__ x,
                                 const float* __restrict__ in_deg,
                                 float* __restrict__ rst) {
    int wave = (int)((blockIdx.x * blockDim.x + threadIdx.x) >> 5);
    int lane = threadIdx.x & 31;
    const int n_tiles = (NN / 16) * 4;   // 25000, divisible by 8 waves/block
    if (wave >= n_tiles) return;         // wave-uniform: EXEC stays all-1s

    int n0    = (wave >> 2) * 16;        // row block
    int j0    = (wave & 3) * 16;         // col block
    int col   = lane & 15;
    int khalf = (lane >> 4) * 2;         // 0 or 2

    const float* arow = h + (long)(n0 + col) * D + khalf;   // A: row = n0 + (lane&15)
    const float* brow = W + (long)(j0 + col) * D + khalf;   // B: W^T column = W row

    v8f acc = {};
#pragma unroll
    for (int kk = 0; kk < 16; ++kk) {
        v2f a = *(const v2f*)(arow + 4 * kk);
        v2f b = *(const v2f*)(brow + 4 * kk);
        acc = __builtin_amdgcn_wmma_f32_16x16x4_f32(
            /*neg_a=*/false, a, /*neg_b=*/false, b,
            /*c_mod=*/(short)0, acc, /*reuse_a=*/false, /*reuse_b=*/false);
    }

    float bj  = bias[j0 + col];
    int   rhi = (lane >> 4) * 8;         // C/D layout: VGPR r holds rows r and r+8
#pragma unroll
    for (int r = 0; r < 8; ++r) {
        int   row = n0 + rhi + r;
        float nrm = rsqrtf(fmaxf(in_deg[row], 1.0f));
        long  off = (long)row * D + j0 + col;
        rst[off] = (acc[r] + bj) * nrm + x[off];
    }
}

// ---------------- kernel 4: e_new = h[src] + h[dst] ----------------------
__global__ void edge_out_kernel(const float* __restrict__ h,
                                const int* __restrict__ src,
                                const int* __restrict__ dst,
                                float* __restrict__ e_new) {
    long tid = (long)blockIdx.x * blockDim.x + threadIdx.x;
    long e   = tid >> 4;
    if (e >= NE) return;
    int q = (int)(tid & 15) * 4;
    int s = src[e];
    int t = dst[e];
    v4f a = *(const v4f*)(h + (long)s * D + q);   // h fits in L2 (25.6 MB)
    v4f b = *(const v4f*)(h + (long)t * D + q);
    v4f r = a + b;
    __builtin_nontemporal_store(r, (v4f*)(e_new + e * (long)D + q)); // streaming out
}

extern "C" void kernel_launch(void* const* d_in, const int* in_sizes, int n_in,
                              void* d_out, int out_size, void* d_ws, size_t ws_size,
                              hipStream_t stream) {
    const float* x        = (const float*)d_in[0];
    const float* edge_fea = (const float*)d_in[1];
    const float* W        = (const float*)d_in[2];
    const float* bias     = (const float*)d_in[3];
    const int*   src      = (const int*)d_in[4];
    const int*   dst      = (const int*)d_in[5];

    float* rst   = (float*)d_out;                  // [NN, D]
    float* e_new = rst + (size_t)NN * D;           // [NE, D]

    float* h      = (float*)d_ws;                  // [NN, D] scratch (25.6 MB)
    float* in_deg = h + (size_t)NN * D;            // [NN]

    // 1) zero h + in_deg (must re-init every call: ws is not re-poisoned)
    int zn = NN * D + NN;
    zero_ws_kernel<<<(zn + 255) / 256, 256, 0, stream>>>(h, zn);

    // 2) scatter-add messages + degree count
    long nt = (long)NE * 16;
    scatter_kernel<<<(int)((nt + 255) / 256), 256, 0, stream>>>(
        x, edge_fea, src, dst, h, in_deg);

    // 3) WMMA GEMM + norm + residual (25000 tiles, 8 waves/block)
    int n_tiles = (NN / 16) * 4;
    gemm_norm_kernel<<<n_tiles / 8, 256, 0, stream>>>(h, W, bias, x, in_deg, rst);

    // 4) edge output
    edge_out_kernel<<<(int)((nt + 255) / 256), 256, 0, stream>>>(h, src, dst, e_new);
}